// HT_CUDA_68977174774313
// MI455X (gfx1250) — compile-verified
//
#include <hip/hip_runtime.h>

// Problem constants (fixed by the reference).
namespace {
constexpr int kB = 2, kL = 1024, kM = 8, kN = 16, kC = 16;
constexpr int kH = 64, kW = 64;
constexpr int kPlane   = kH * kW;        // 4096 bins
constexpr int kHist    = kC * kPlane;    // 65536 floats = 256 KB LDS
constexpr int kVotes   = kM * kN;        // 128 votes per anchor
constexpr int kThreads = 1024;           // 32 wave32s
}

// Raw clang vector types (usable in builtins on both host and device passes).
typedef int   v4i __attribute__((vector_size(16)));
typedef float v4f __attribute__((vector_size(16)));
// Async-LDS builtin signature (revealed by compiler diagnostics):
//   void __builtin_amdgcn_global_store_async_from_lds_b128(v4i AS1*, v4i AS3*, Ii offset, Ii cpol)
typedef __attribute__((address_space(1))) v4i* gptr_v4i;
typedef __attribute__((address_space(3))) v4i* lptr_v4i;

__global__ __launch_bounds__(kThreads)
void ht_vote_kernel(const float* __restrict__ feats,       // [B,L,N,C]
                    const int*   __restrict__ voxels_src,  // [B,L,2]
                    const int*   __restrict__ voxels_dst,  // [B,L,2]
                    const int*   __restrict__ idxs_src,    // [B,L,M]
                    const int*   __restrict__ idxs_dst,    // [B,L,N]
                    float*       __restrict__ out)         // [B,L,C,H,W]
{
  __shared__ float hist[kHist];          // full per-anchor histogram in LDS (CDNA5: 320KB/WGP)
  __shared__ int   vote_bin[kVotes];     // y*W+x or -1 if out of range
  __shared__ int   vote_src[kVotes];     // gathered src index s

  const int bl  = blockIdx.x;            // b*L + l
  const int b   = bl >> 10;              // / kL (L == 1024)
  const int tid = threadIdx.x;

  // ---- zero the 256 KB LDS histogram (128-bit wide) ----
  v4f z = {0.0f, 0.0f, 0.0f, 0.0f};
  #pragma unroll
  for (int i = tid; i < kHist / 4; i += kThreads)
    ((v4f*)hist)[i] = z;

  // ---- stage per-vote metadata (integer math: floor(vdst - vsrc) is exact) ----
  if (tid < kVotes) {
    const int m = tid >> 4;              // / kN
    const int n = tid & (kN - 1);
    const int s = idxs_src[(size_t)bl * kM + m];
    const int d = idxs_dst[((size_t)b * kL + s) * kN + n];
    const int by = voxels_dst[((size_t)b * kL + d) * 2 + 0]
                 - voxels_src[((size_t)b * kL + s) * 2 + 0] + kH / 2;
    const int bx = voxels_dst[((size_t)b * kL + d) * 2 + 1]
                 - voxels_src[((size_t)b * kL + s) * 2 + 1] + kW / 2;
    const bool ok = (by >= 0) && (by < kH) && (bx >= 0) && (bx < kW);
    vote_bin[tid] = ok ? (by * kW + bx) : -1;
    vote_src[tid] = s;
  }
  __syncthreads();

  // ---- scatter votes with LDS float atomics (ds_add_f32) ----
  // Lane mapping: v (vote) varies fastest across lanes so concurrent lanes hit
  // different random bins (good bank spread); c stride is 4096 dwords which
  // would alias to one bank if c varied across lanes instead.
  for (int p = tid; p < kVotes * kC; p += kThreads) {
    const int v   = p & (kVotes - 1);
    const int c   = p >> 7;              // / kVotes
    const int bin = vote_bin[v];
    if (bin >= 0) {
      const int s = vote_src[v];
      const int n = v & (kN - 1);
      const float w = feats[(((size_t)b * kL + s) * kN + n) * kC + c];
      atomicAdd(&hist[c * kPlane + bin], w);
    }
  }
  __syncthreads();

  // ---- stream LDS -> global: 256 KB contiguous, already in [c][h][w] order ----
  float* dst = out + (size_t)bl * kHist;

#if defined(__gfx1250__) && __has_builtin(__builtin_amdgcn_global_store_async_from_lds_b128)
  // CDNA5 async LDS->memory path (ASYNCcnt-tracked, 128b per lane).
  // The instruction's 24-bit immediate offset is added to BOTH the LDS and the
  // global address (CDNA5 ISA), so 16 back-to-back async stores
  // with constant offsets stream the whole 256 KB with zero per-step VALU work.
  {
    float* gbase = dst  + tid * 4;       // 16 B per lane, 512 B per wave step
    float* lbase = hist + tid * 4;
    #define HT_ASYNC_STEP(J)                                              \
      __builtin_amdgcn_global_store_async_from_lds_b128(                  \
          (gptr_v4i)gbase, (lptr_v4i)lbase,                               \
          /*offset=*/(J) * (kThreads * 16), /*cpol=*/0)
    HT_ASYNC_STEP(0);  HT_ASYNC_STEP(1);  HT_ASYNC_STEP(2);  HT_ASYNC_STEP(3);
    HT_ASYNC_STEP(4);  HT_ASYNC_STEP(5);  HT_ASYNC_STEP(6);  HT_ASYNC_STEP(7);
    HT_ASYNC_STEP(8);  HT_ASYNC_STEP(9);  HT_ASYNC_STEP(10); HT_ASYNC_STEP(11);
    HT_ASYNC_STEP(12); HT_ASYNC_STEP(13); HT_ASYNC_STEP(14); HT_ASYNC_STEP(15);
    #undef HT_ASYNC_STEP
  }
  #if __has_builtin(__builtin_amdgcn_s_wait_asynccnt)
  __builtin_amdgcn_s_wait_asynccnt(0);   // (S_ENDPGM also implies wait-idle)
  #endif
#else
  // Fallback: wide non-temporal stores (output is write-once, streaming).
  for (int i = tid * 4; i < kHist; i += kThreads * 4) {
    v4f vld = *(const v4f*)(hist + i);
    __builtin_nontemporal_store(vld, (v4f*)(dst + i));
  }
#endif
}

extern "C" void kernel_launch(void* const* d_in, const int* in_sizes, int n_in,
                              void* d_out, int out_size, void* d_ws, size_t ws_size,
                              hipStream_t stream) {
  const float* feats      = (const float*)d_in[0];  // [B,L,N,C] fp32
  const int*   voxels_src = (const int*)d_in[1];    // [B,L,2]
  const int*   voxels_dst = (const int*)d_in[2];    // [B,L,2]
  const int*   idxs_src   = (const int*)d_in[3];    // [B,L,M]
  const int*   idxs_dst   = (const int*)d_in[4];    // [B,L,N]
  float*       out        = (float*)d_out;          // [B,L,C,H,W] fp32

  dim3 grid(kB * kL);       // one workgroup per anchor
  dim3 block(kThreads);
  ht_vote_kernel<<<grid, block, 0, stream>>>(feats, voxels_src, voxels_dst,
                                             idxs_src, idxs_dst, out);
}